// InputModule_68272800137538
// MI455X (gfx1250) — compile-verified
//
#include <hip/hip_runtime.h>
#include <math.h>

// ---------------------------------------------------------------------------
// Problem constants (match reference)
// ---------------------------------------------------------------------------
#define BATCH 32
#define QLEN  32
#define DOCL  2048
#define NSENT 64
#define EMB   300
#define HID   256
#define G3    768   // 3*HID

typedef __attribute__((ext_vector_type(2))) float v2f;
typedef __attribute__((ext_vector_type(8))) float v8f;

// D = A(16x4) * B(4x16) + C(16x16), all f32, wave32.
__device__ __forceinline__ v8f wmma_f32(v2f a, v2f b, v8f c) {
    return __builtin_amdgcn_wmma_f32_16x16x4_f32(
        /*neg_a=*/false, a, /*neg_b=*/false, b,
        /*c_mod=*/(short)0, c, /*reuse_a=*/false, /*reuse_b=*/false);
}

__device__ __forceinline__ float sigmoidf_(float x) {
    return 1.0f / (1.0f + __expf(-x));
}

// ---------------------------------------------------------------------------
// facts[b][s][:] = mean over sentence-s tokens of emb[docs[b][tok]]
// One block per (b, s). Segment boundaries = prefix sums of sentence_lengths.
// ---------------------------------------------------------------------------
__global__ void facts_kernel(const float* __restrict__ emb,
                             const int*   __restrict__ docs,
                             const int*   __restrict__ slen,
                             float*       __restrict__ facts) {
    int b = blockIdx.x / NSENT;
    int s = blockIdx.x % NSENT;
    const int* lb = slen + b * NSENT;
    int start = 0;
    for (int i = 0; i < s; ++i) start += lb[i];
    int len = lb[s];
    if (start > DOCL) start = DOCL;
    if (start + len > DOCL) len = DOCL - start;
    if (len < 0) len = 0;
    float inv = 1.0f / (float)(len > 0 ? len : 1);
    const int* db = docs + b * DOCL;
    for (int e = threadIdx.x; e < EMB; e += blockDim.x) {
        float sum = 0.0f;
        for (int j = 0; j < len; ++j)
            sum += emb[(size_t)db[start + j] * EMB + e];
        facts[((size_t)b * NSENT + s) * EMB + e] = sum * inv;
    }
}

// ---------------------------------------------------------------------------
// xg[row][g] = sum_k A[row][k] * wih[g][k] + bih[g]
//   A row = emb[idx[row]] (fused gather) if idx != nullptr, else x[row].
//   M = Mtiles*16 rows, N = 768, K = 300 (multiple of 4).
// One wave per 16x64 output slab: 4 adjacent 16x16 N-tiles so each A
// fragment feeds 4 WMMAs (4x A-side load reuse).
// ---------------------------------------------------------------------------
#define NTPW 4                       // N-tiles per wave
#define NGRP (G3 / (16 * NTPW))      // 12 column groups

__global__ void __launch_bounds__(256)
proj_gemm_kernel(const float* __restrict__ emb,
                 const int*   __restrict__ idx,
                 const float* __restrict__ x,
                 const float* __restrict__ wih,
                 const float* __restrict__ bih,
                 float*       __restrict__ xg,
                 int Mtiles) {
    const int gwave = (blockIdx.x * blockDim.x + threadIdx.x) >> 5;
    const int lane  = threadIdx.x & 31;
    const int grp   = gwave % NGRP;
    const int mtile = gwave / NGRP;
    if (mtile >= Mtiles) return;                 // wave-uniform exit

    const int khalf = (lane >> 4) * 2;           // A/B fragment K offset
    const int mrow  = mtile * 16 + (lane & 15);  // A fragment row
    const int g0    = grp * (16 * NTPW) + (lane & 15);

    const float* arow = idx ? (emb + (size_t)idx[mrow] * EMB)
                            : (x   + (size_t)mrow      * EMB);
    const float* brow[NTPW];
    for (int j = 0; j < NTPW; ++j)               // B[k][g] = wih[g][k]
        brow[j] = wih + (size_t)(g0 + 16 * j) * EMB;

    v8f acc[NTPW] = {};
    for (int k = 0; k < EMB; k += 4) {
        v2f a;
        a.x = arow[k + khalf + 0];
        a.y = arow[k + khalf + 1];
        for (int j = 0; j < NTPW; ++j) {
            v2f bf;
            bf.x = brow[j][k + khalf + 0];
            bf.y = brow[j][k + khalf + 1];
            acc[j] = wmma_f32(a, bf, acc[j]);
        }
    }

    const int mbase = mtile * 16 + ((lane >> 4) * 8);  // D row base
    for (int j = 0; j < NTPW; ++j) {
        const int col = g0 + 16 * j;
        const float bv = bih[col];
        for (int v = 0; v < 8; ++v) {
            int row = mbase + v;
            xg[(size_t)row * G3 + col] = acc[j][v] + bv;
        }
    }
}

// ---------------------------------------------------------------------------
// Persistent single-workgroup GRU scan.
//   xg : (BATCH, T, 768) precomputed input projections (+bih)
//   whh: (768, 256), bhh: (768)
//   mode 0: query GRU  -> out[b*256+col] = h_t at t == qlen[b]-1
//   mode 1: fusion fwd -> out[(b*T+t)*256+col]  = h_t
//   mode 2: fusion bwd -> out[(b*T+t)*256+col] += h_t (t runs T-1..0)
// 32 waves: wave w owns mtile = w>>4 (batch rows), coltile = w&15 (H cols),
// and computes the r/z/n gate tiles for those columns with 3 WMMA chains
// sharing one A fragment. h lives in LDS (32 KB); one barrier pair per step.
// ---------------------------------------------------------------------------
__global__ void __launch_bounds__(1024)
gru_scan_kernel(const float* __restrict__ xg,
                const float* __restrict__ whh,
                const float* __restrict__ bhh,
                const int*   __restrict__ qlen,
                float*       __restrict__ out,
                int T, int mode) {
    __shared__ float hbuf[BATCH][HID];

    const int tid  = threadIdx.x;
    const int wave = tid >> 5;
    const int lane = tid & 31;
    const int mt   = wave >> 4;                  // 0..1  (batch tile)
    const int nt   = wave & 15;                  // 0..15 (column tile)
    const int khalf = (lane >> 4) * 2;
    const int arow  = mt * 16 + (lane & 15);     // batch row for A loads
    const int ncol  = nt * 16 + (lane & 15);     // H column (0..255)
    const int mbase = mt * 16 + ((lane >> 4) * 8);

    for (int i = tid; i < BATCH * HID; i += 1024)
        (&hbuf[0][0])[i] = 0.0f;
    __syncthreads();

    // B[k][g] = whh[g][k]; gate g for this lane's column:
    const float* wr = whh + (size_t)(ncol        ) * HID;
    const float* wz = whh + (size_t)(ncol + HID  ) * HID;
    const float* wn = whh + (size_t)(ncol + 2*HID) * HID;
    const float br = bhh[ncol], bz = bhh[ncol + HID], bn = bhh[ncol + 2*HID];

    for (int step = 0; step < T; ++step) {
        const int t = (mode == 2) ? (T - 1 - step) : step;

        v8f dr = {}, dz = {}, dn = {};
        for (int k = 0; k < HID; k += 4) {
            v2f a, b0, b1, b2;
            a.x  = hbuf[arow][k + khalf + 0];
            a.y  = hbuf[arow][k + khalf + 1];
            b0.x = wr[k + khalf + 0]; b0.y = wr[k + khalf + 1];
            b1.x = wz[k + khalf + 0]; b1.y = wz[k + khalf + 1];
            b2.x = wn[k + khalf + 0]; b2.y = wn[k + khalf + 1];
            dr = wmma_f32(a, b0, dr);
            dz = wmma_f32(a, b1, dz);
            dn = wmma_f32(a, b2, dn);
        }

        float hnew[8];
        for (int v = 0; v < 8; ++v) {
            const int bi = mbase + v;            // batch index
            const float* xrow = xg + ((size_t)bi * T + t) * G3;
            float ir  = xrow[ncol];
            float iz  = xrow[ncol + HID];
            float inn = xrow[ncol + 2*HID];
            float hr = dr[v] + br;
            float hz = dz[v] + bz;
            float hn = dn[v] + bn;
            float r = sigmoidf_(ir + hr);
            float z = sigmoidf_(iz + hz);
            float n = tanhf(inn + r * hn);
            float hp = hbuf[bi][ncol];
            hnew[v] = (1.0f - z) * n + z * hp;
        }

        __syncthreads();   // all h reads (WMMA A + hp) complete
        for (int v = 0; v < 8; ++v) {
            const int bi = mbase + v;
            hbuf[bi][ncol] = hnew[v];
            if (mode == 0) {
                if (qlen[bi] - 1 == t)
                    out[(size_t)bi * HID + ncol] = hnew[v];
            } else if (mode == 1) {
                out[((size_t)bi * T + t) * HID + ncol] = hnew[v];
            } else {
                out[((size_t)bi * T + t) * HID + ncol] += hnew[v];
            }
        }
        __syncthreads();   // h updated before next step's reads
    }
}

// num_facts[b] = count(sentence_lengths[b][s] > 0), emitted as float.
__global__ void numfacts_kernel(const int* __restrict__ slen,
                                float* __restrict__ out) {
    int b = threadIdx.x;
    if (b < BATCH) {
        int c = 0;
        for (int s = 0; s < NSENT; ++s) c += (slen[b * NSENT + s] > 0);
        out[b] = (float)c;
    }
}

// ---------------------------------------------------------------------------
extern "C" void kernel_launch(void* const* d_in, const int* in_sizes, int n_in,
                              void* d_out, int out_size, void* d_ws, size_t ws_size,
                              hipStream_t stream) {
    (void)in_sizes; (void)n_in; (void)out_size; (void)ws_size;

    const int*   queries = (const int*)  d_in[0];
    const int*   qlen    = (const int*)  d_in[1];
    const int*   docs    = (const int*)  d_in[2];
    const int*   slen    = (const int*)  d_in[3];
    const float* emb     = (const float*)d_in[4];
    const float* q_wih   = (const float*)d_in[5];
    const float* q_whh   = (const float*)d_in[6];
    const float* q_bih   = (const float*)d_in[7];
    const float* q_bhh   = (const float*)d_in[8];
    const float* f_wih   = (const float*)d_in[9];
    const float* f_whh   = (const float*)d_in[10];
    const float* f_bih   = (const float*)d_in[11];
    const float* f_bhh   = (const float*)d_in[12];
    const float* b_wih   = (const float*)d_in[13];
    const float* b_whh   = (const float*)d_in[14];
    const float* b_bih   = (const float*)d_in[15];
    const float* b_bhh   = (const float*)d_in[16];

    float* out   = (float*)d_out;
    float* qrep  = out;                                     // 32*256
    float* fout  = out + BATCH * HID;                       // 32*64*256
    float* nfout = out + BATCH * HID + BATCH * NSENT * HID; // 32

    float* ws    = (float*)d_ws;
    float* facts = ws;                                    // 2048*300
    float* xg_q  = facts + (size_t)BATCH * NSENT * EMB;   // 1024*768
    float* xg_f  = xg_q  + (size_t)BATCH * QLEN * G3;     // 2048*768
    float* xg_b  = xg_f  + (size_t)BATCH * NSENT * G3;    // 2048*768

    // 1) per-sentence mean pooling (ragged gather + segment mean)
    facts_kernel<<<BATCH * NSENT, 128, 0, stream>>>(emb, docs, slen, facts);

    // 2) input projections (WMMA f32 GEMMs); query gather fused into GEMM
    {
        const int mtq = (BATCH * QLEN) / 16;   // 64
        const int mtf = (BATCH * NSENT) / 16;  // 128
        proj_gemm_kernel<<<(mtq * NGRP + 7) / 8, 256, 0, stream>>>(
            emb, queries, nullptr, q_wih, q_bih, xg_q, mtq);
        proj_gemm_kernel<<<(mtf * NGRP + 7) / 8, 256, 0, stream>>>(
            emb, nullptr, facts, f_wih, f_bih, xg_f, mtf);
        proj_gemm_kernel<<<(mtf * NGRP + 7) / 8, 256, 0, stream>>>(
            emb, nullptr, facts, b_wih, b_bih, xg_b, mtf);
    }

    // 3) recurrences: persistent single-workgroup scans (h in LDS)
    gru_scan_kernel<<<1, 1024, 0, stream>>>(xg_q, q_whh, q_bhh, qlen, qrep,
                                            QLEN, /*mode=*/0);
    gru_scan_kernel<<<1, 1024, 0, stream>>>(xg_f, f_whh, f_bhh, nullptr, fout,
                                            NSENT, /*mode=*/1);
    gru_scan_kernel<<<1, 1024, 0, stream>>>(xg_b, b_whh, b_bhh, nullptr, fout,
                                            NSENT, /*mode=*/2);

    // 4) num_facts
    numfacts_kernel<<<1, 32, 0, stream>>>(slen, nfout);
}